// SparseCloudConvolution_11184094839595
// MI455X (gfx1250) — compile-verified
//
#include <hip/hip_runtime.h>

#define N_NODES 20000
#define N_EDGES 640000
#define C_IN    128
#define K_FOLDS 8
#define FILTERS 128
#define ZROW    (K_FOLDS * FILTERS)   // 1024 floats per node in Z

typedef float v2f __attribute__((ext_vector_type(2)));
typedef float v8f __attribute__((ext_vector_type(8)));

// ---------------------------------------------------------------------------
// Kernel 0: out[n][f] = bias[f]   (harness poisons d_out; rebuild every call)
// ---------------------------------------------------------------------------
__global__ void init_out_kernel(const float* __restrict__ bias,
                                float* __restrict__ out) {
    int i = blockIdx.x * blockDim.x + threadIdx.x;
    if (i < N_NODES * FILTERS) out[i] = bias[i & (FILTERS - 1)];
}

// ---------------------------------------------------------------------------
// Kernel 1: Z[n][k][f] = sum_c X[n][c] * W[k][c][f]
// One wave computes a 16(M) x 64(N) strip of one fold using
// V_WMMA_F32_16X16X4_F32 (fp32 A/B, fp32 acc) -- exact fp32 math.
// Waves: 1250 row-tiles * 2 N-halves * 8 folds = 20000 -> 2500 blocks x 256.
// ---------------------------------------------------------------------------
__global__ __launch_bounds__(256) void gemm_z_kernel(
        const float* __restrict__ X,     // (N_NODES, C_IN) row-major
        const float* __restrict__ Wall,  // (K_FOLDS, C_IN, FILTERS)
        float* __restrict__ Z) {         // (N_NODES, K_FOLDS, FILTERS)
    const int lane = threadIdx.x & 31;
    const int wid  = blockIdx.x * 8 + (threadIdx.x >> 5);  // 0..19999
    const int fold = wid & 7;
    const int nh   = (wid >> 3) & 1;
    const int rt   = wid >> 4;          // 0..1249

    const int m0 = rt * 16;
    const int n0 = nh * 64;
    const int m  = lane & 15;           // A: row within tile
    const int kh = lane >> 4;           // A/B: lanes 16-31 hold K=2,3

    const float* W    = Wall + (size_t)fold * (C_IN * FILTERS);
    const float* xrow = X + (size_t)(m0 + m) * C_IN;

    v8f acc0 = {}, acc1 = {}, acc2 = {}, acc3 = {};

    for (int kb = 0; kb < C_IN; kb += 4) {
        const int k0 = kb + kh * 2;
        // A tile 16x4 f32: vgpr0 = K=k0, vgpr1 = K=k0+1 (per lane half)
        v2f a;
        a.x = xrow[k0];
        a.y = xrow[k0 + 1];
        // B tile 4x16 f32: vgpr0 = row k0, vgpr1 = row k0+1, N = lane&15
        const float* w0 = W + (size_t)k0 * FILTERS + n0 + (lane & 15);
        const float* w1 = w0 + FILTERS;
        v2f b0, b1, b2, b3;
        b0.x = w0[0];  b0.y = w1[0];
        b1.x = w0[16]; b1.y = w1[16];
        b2.x = w0[32]; b2.y = w1[32];
        b3.x = w0[48]; b3.y = w1[48];
        acc0 = __builtin_amdgcn_wmma_f32_16x16x4_f32(false, a, false, b0, (short)0, acc0, false, false);
        acc1 = __builtin_amdgcn_wmma_f32_16x16x4_f32(false, a, false, b1, (short)0, acc1, false, false);
        acc2 = __builtin_amdgcn_wmma_f32_16x16x4_f32(false, a, false, b2, (short)0, acc2, false, false);
        acc3 = __builtin_amdgcn_wmma_f32_16x16x4_f32(false, a, false, b3, (short)0, acc3, false, false);
    }

    // C/D layout: vgpr r holds M = (lane>>4)*8 + r, N = lane&15
    const int mrow = m0 + kh * 8;
    const int ncol = n0 + (lane & 15);
    #pragma unroll
    for (int r = 0; r < 8; ++r) {
        float* zr = Z + (size_t)(mrow + r) * ZROW + fold * FILTERS;
        zr[ncol]      = acc0[r];
        zr[ncol + 16] = acc1[r];
        zr[ncol + 32] = acc2[r];
        zr[ncol + 48] = acc3[r];
    }
}

// ---------------------------------------------------------------------------
// Kernel 2: per edge e: out[row[e]] += sum_k ef[k][e] * Z[col[e]][k][:]
// One wave per edge; lane l owns filters 4l..4l+3 (float4).
// Z row (4KB, contiguous) is L2-resident (82MB < 192MB L2).
// ---------------------------------------------------------------------------
__global__ __launch_bounds__(256) void edge_scatter_kernel(
        const float* __restrict__ ef,    // (K_FOLDS, N_EDGES)
        const int*   __restrict__ idx,   // (N_EDGES, 2): row, col
        const float* __restrict__ Z,     // (N_NODES, K_FOLDS, FILTERS)
        float* __restrict__ out) {       // (N_NODES, FILTERS)
    const int lane = threadIdx.x & 31;
    const int e    = blockIdx.x * 8 + (threadIdx.x >> 5);  // exact cover

    const int row = idx[2 * e];
    const int col = idx[2 * e + 1];

    const float4* zp = (const float4*)(Z + (size_t)col * ZROW);
    float4 acc = {0.f, 0.f, 0.f, 0.f};
    #pragma unroll
    for (int k = 0; k < K_FOLDS; ++k) {
        const float w = ef[(size_t)k * N_EDGES + e];  // uniform per wave
        const float4 z = zp[k * 32 + lane];
        acc.x += w * z.x;
        acc.y += w * z.y;
        acc.z += w * z.z;
        acc.w += w * z.w;
    }

    float* o = out + (size_t)row * FILTERS + lane * 4;
    unsafeAtomicAdd(o + 0, acc.x);
    unsafeAtomicAdd(o + 1, acc.y);
    unsafeAtomicAdd(o + 2, acc.z);
    unsafeAtomicAdd(o + 3, acc.w);
}

// ---------------------------------------------------------------------------
extern "C" void kernel_launch(void* const* d_in, const int* in_sizes, int n_in,
                              void* d_out, int out_size, void* d_ws, size_t ws_size,
                              hipStream_t stream) {
    const float* X    = (const float*)d_in[0];  // node_features
    const float* EF   = (const float*)d_in[1];  // edge_features (K, E)
    const float* Wall = (const float*)d_in[2];  // kernel (K, C, F)
    const float* bias = (const float*)d_in[3];  // bias (F)
    const int*   idx  = (const int*)  d_in[4];  // indices (E, 2)
    float* out = (float*)d_out;
    float* Z   = (float*)d_ws;                  // needs 20000*8*128*4 = 81.92 MB

    // 0: out = bias (broadcast)
    init_out_kernel<<<(N_NODES * FILTERS + 255) / 256, 256, 0, stream>>>(bias, out);

    // 1: Z = X @ W_k for all folds (fp32 WMMA)
    gemm_z_kernel<<<2500, 256, 0, stream>>>(X, Wall, Z);

    // 2: gather-weight-scatter over edges
    edge_scatter_kernel<<<N_EDGES / 8, 256, 0, stream>>>(EF, idx, Z, out);
}